// GatFCM_74302934220972
// MI455X (gfx1250) — compile-verified
//
#include <hip/hip_runtime.h>

#define NN   50000
#define DIN  1028
#define DOUT 256
#define NEG_SLOPE 0.2f

typedef float v2f __attribute__((ext_vector_type(2)));
typedef float v8f __attribute__((ext_vector_type(8)));

// ---------------------------------------------------------------------------
// Init: zero the output accumulator, emax = -inf, denom = 0
// ---------------------------------------------------------------------------
__global__ void __launch_bounds__(256)
gat_init(float* __restrict__ out, float* __restrict__ emax, float* __restrict__ denom) {
  const int i = blockIdx.x * blockDim.x + threadIdx.x;
  if (i < NN * DOUT) out[i] = 0.0f;
  if (i < NN) { emax[i] = -__builtin_inff(); denom[i] = 0.0f; }
}

// ---------------------------------------------------------------------------
// GEMM via V_WMMA_F32_16X16X4_F32.
// Block = 128 threads (4 waves). Block computes a 16(M) x 256(N) tile of
// out = x @ W.  Wave w owns columns [w*64, w*64+64) as four 16x16 WMMA tiles.
// grid.z = 0 -> xl = x@Wl ; grid.z = 1 -> xr = x@Wr.
// A-frag (16x4 f32): lanes 0-15 row M=l15 K={k,k+1}; lanes 16-31 K={k+2,k+3}.
// B-frag (4x16 f32): VGPR0 lanes0-15 = row K (N=lane), lanes16-31 = row K+2.
// C/D (16x16 f32):  VGPR v, lanes0-15 -> M=v, lanes16-31 -> M=v+8.
// ---------------------------------------------------------------------------
__global__ void __launch_bounds__(128)
gat_gemm(const float* __restrict__ x, const float* __restrict__ Wl,
         const float* __restrict__ Wr, float* __restrict__ xl, float* __restrict__ xr) {
  const float* __restrict__ W   = (blockIdx.z == 0) ? Wl : Wr;
  float* __restrict__       out = (blockIdx.z == 0) ? xl : xr;

  const int m0   = blockIdx.x * 16;
  const int wave = threadIdx.x >> 5;
  const int lane = threadIdx.x & 31;
  const int half = lane >> 4;        // 0: lanes 0-15, 1: lanes 16-31
  const int l15  = lane & 15;
  const int nc   = wave * 64 + l15;  // this lane's N column within its 4 tiles

  v8f acc0 = {}, acc1 = {}, acc2 = {}, acc3 = {};
  const float* __restrict__ arow = x + (size_t)(m0 + l15) * DIN;

  for (int k = 0; k < DIN; k += 4) {          // 257 exact K-steps
    const int kk = k + 2 * half;
    v2f a = *(const v2f*)(arow + kk);         // contiguous 8B load
    const float* __restrict__ wp = W + (size_t)kk * DOUT + nc;
    v2f b0, b1, b2, b3;
    b0.x = wp[0];  b0.y = wp[DOUT + 0];
    b1.x = wp[16]; b1.y = wp[DOUT + 16];
    b2.x = wp[32]; b2.y = wp[DOUT + 32];
    b3.x = wp[48]; b3.y = wp[DOUT + 48];
    acc0 = __builtin_amdgcn_wmma_f32_16x16x4_f32(false, a, false, b0, (short)0, acc0, false, false);
    acc1 = __builtin_amdgcn_wmma_f32_16x16x4_f32(false, a, false, b1, (short)0, acc1, false, false);
    acc2 = __builtin_amdgcn_wmma_f32_16x16x4_f32(false, a, false, b2, (short)0, acc2, false, false);
    acc3 = __builtin_amdgcn_wmma_f32_16x16x4_f32(false, a, false, b3, (short)0, acc3, false, false);
  }

#pragma unroll
  for (int v = 0; v < 8; ++v) {
    const size_t row = (size_t)(m0 + v + half * 8) * DOUT;
    out[row + nc +  0] = acc0[v];
    out[row + nc + 16] = acc1[v];
    out[row + nc + 32] = acc2[v];
    out[row + nc + 48] = acc3[v];
  }
}

// ---------------------------------------------------------------------------
// Float segment-max via int atomics (emax pre-initialized to -inf).
// ---------------------------------------------------------------------------
__device__ __forceinline__ void atomicMaxF(float* addr, float val) {
  if (val >= 0.0f) atomicMax((int*)addr, __float_as_int(val));
  else             atomicMin((unsigned int*)addr, __float_as_uint(val));
}

// ---------------------------------------------------------------------------
// Edge logits: one wave32 per edge. e = att . leakyrelu(xl[src] + xr[dst]);
// segment-max into emax[dst]. Self-loops appended at index >= nE.
// ---------------------------------------------------------------------------
__global__ void __launch_bounds__(256)
gat_edge_e(const int* __restrict__ srcs, const int* __restrict__ dsts,
           const float* __restrict__ xl, const float* __restrict__ xr,
           const float* __restrict__ att, float* __restrict__ e,
           float* __restrict__ emax, int nE, int total) {
  const int edge = (int)((blockIdx.x * (unsigned)blockDim.x + threadIdx.x) >> 5);
  const int lane = threadIdx.x & 31;
  if (edge >= total) return;
  int s, d;
  if (edge < nE) { s = srcs[edge]; d = dsts[edge]; } else { s = d = edge - nE; }

  const float4* __restrict__ pa = (const float4*)(xl + (size_t)s * DOUT) + lane * 2;
  const float4* __restrict__ pb = (const float4*)(xr + (size_t)d * DOUT) + lane * 2;
  const float4* __restrict__ pw = (const float4*)att + lane * 2;

  float sum = 0.0f;
#pragma unroll
  for (int j = 0; j < 2; ++j) {
    float4 a = pa[j], b = pb[j], w = pw[j];
    float v0 = a.x + b.x, v1 = a.y + b.y, v2 = a.z + b.z, v3 = a.w + b.w;
    v0 = (v0 > 0.0f) ? v0 : v0 * NEG_SLOPE;
    v1 = (v1 > 0.0f) ? v1 : v1 * NEG_SLOPE;
    v2 = (v2 > 0.0f) ? v2 : v2 * NEG_SLOPE;
    v3 = (v3 > 0.0f) ? v3 : v3 * NEG_SLOPE;
    sum += v0 * w.x + v1 * w.y + v2 * w.z + v3 * w.w;
  }
#pragma unroll
  for (int off = 16; off > 0; off >>= 1) sum += __shfl_xor(sum, off, 32);
  if (lane == 0) { e[edge] = sum; atomicMaxF(emax + d, sum); }
}

// ---------------------------------------------------------------------------
// p = exp(e - emax[dst]); denom[dst] += p  (one thread per edge)
// ---------------------------------------------------------------------------
__global__ void __launch_bounds__(256)
gat_edge_p(const int* __restrict__ dsts, float* __restrict__ e,
           const float* __restrict__ emax, float* __restrict__ denom,
           int nE, int total) {
  const int i = blockIdx.x * blockDim.x + threadIdx.x;
  if (i >= total) return;
  const int d = (i < nE) ? dsts[i] : (i - nE);
  const float p = __expf(e[i] - emax[d]);
  e[i] = p;
  atomicAdd(denom + d, p);
}

// ---------------------------------------------------------------------------
// out[dst] += (p/denom[dst]) * xl[src]  (one wave32 per edge, 8 floats/lane)
// ---------------------------------------------------------------------------
__global__ void __launch_bounds__(256)
gat_scatter(const int* __restrict__ srcs, const int* __restrict__ dsts,
            const float* __restrict__ xl, const float* __restrict__ e,
            const float* __restrict__ denom, float* __restrict__ out,
            int nE, int total) {
  const int edge = (int)((blockIdx.x * (unsigned)blockDim.x + threadIdx.x) >> 5);
  const int lane = threadIdx.x & 31;
  if (edge >= total) return;
  int s, d;
  if (edge < nE) { s = srcs[edge]; d = dsts[edge]; } else { s = d = edge - nE; }

  const float alpha = e[edge] / denom[d];
  const float4* __restrict__ pa = (const float4*)(xl + (size_t)s * DOUT) + lane * 2;
  float* __restrict__ po = out + (size_t)d * DOUT + lane * 8;
#pragma unroll
  for (int j = 0; j < 2; ++j) {
    float4 a = pa[j];
    atomicAdd(po + 4 * j + 0, alpha * a.x);
    atomicAdd(po + 4 * j + 1, alpha * a.y);
    atomicAdd(po + 4 * j + 2, alpha * a.z);
    atomicAdd(po + 4 * j + 3, alpha * a.w);
  }
}

// ---------------------------------------------------------------------------
// out += bias (broadcast over rows), in place
// ---------------------------------------------------------------------------
__global__ void __launch_bounds__(256)
gat_bias(float* __restrict__ out, const float* __restrict__ bias) {
  const int i = blockIdx.x * blockDim.x + threadIdx.x;
  if (i < NN * DOUT) out[i] += bias[i & (DOUT - 1)];
}

// ---------------------------------------------------------------------------
extern "C" void kernel_launch(void* const* d_in, const int* in_sizes, int n_in,
                              void* d_out, int out_size, void* d_ws, size_t ws_size,
                              hipStream_t stream) {
  (void)n_in; (void)out_size; (void)ws_size;
  const float* x    = (const float*)d_in[0];
  const int*   ei   = (const int*)d_in[1];   // [2, E] row-major: src row, dst row
  const float* Wl   = (const float*)d_in[2];
  const float* Wr   = (const float*)d_in[3];
  const float* att  = (const float*)d_in[4];
  const float* bias = (const float*)d_in[5];
  float*       out  = (float*)d_out;

  const int E     = in_sizes[1] / 2;
  const int total = E + NN;
  const int* srcs = ei;
  const int* dsts = ei + E;

  // Workspace layout (floats): xl | xr | e(total) | emax | denom  (~110 MB)
  float* xl    = (float*)d_ws;
  float* xr    = xl + (size_t)NN * DOUT;
  float* e     = xr + (size_t)NN * DOUT;
  float* emax  = e + total;
  float* denom = emax + NN;

  // 1) init accumulators
  {
    int n = NN * DOUT;
    gat_init<<<(n + 255) / 256, 256, 0, stream>>>(out, emax, denom);
  }
  // 2) xl = x@Wl, xr = x@Wr  (WMMA f32 16x16x4; exact tiling, no remainder)
  {
    dim3 grid(NN / 16, 1, 2);
    gat_gemm<<<grid, 128, 0, stream>>>(x, Wl, Wr, xl, xr);
  }
  // 3) edge logits + segment max (one wave per edge)
  {
    long long threads = (long long)total * 32;
    gat_edge_e<<<(unsigned)((threads + 255) / 256), 256, 0, stream>>>(
        srcs, dsts, xl, xr, att, e, emax, E, total);
  }
  // 4) exp + segment sum
  gat_edge_p<<<(total + 255) / 256, 256, 0, stream>>>(dsts, e, emax, denom, E, total);
  // 5) weighted scatter into out
  {
    long long threads = (long long)total * 32;
    gat_scatter<<<(unsigned)((threads + 255) / 256), 256, 0, stream>>>(
        srcs, dsts, xl, e, denom, out, E, total);
  }
  // 6) bias
  gat_bias<<<(NN * DOUT + 255) / 256, 256, 0, stream>>>(out, bias);
}